// SurgiATM_11879879544451
// MI455X (gfx1250) — compile-verified
//
#include <hip/hip_runtime.h>
#include <hip/hip_bf16.h>

#define HH 512
#define WW 512
#define HWSZ (HH*WW)          // 262144 = 2^18
#define NBATCH 16
#define IMG3 (3*HWSZ)
#define ETA_F 0.1f
#define INV225 (1.0f/225.0f)
#define EPS_F 1e-3f

typedef float v2f __attribute__((ext_vector_type(2)));
typedef float v8f __attribute__((ext_vector_type(8)));

#if defined(__gfx1250__) && __has_builtin(__builtin_amdgcn_wmma_f32_16x16x4_f32)
#define USE_WMMA 1
#else
#define USE_WMMA 0
#endif

#if defined(__gfx1250__) && __has_builtin(__builtin_amdgcn_tensor_load_to_lds)
#define HAVE_TDM 1
#else
#define HAVE_TDM 0
#endif

// TDM-compatible padded LDS layout: rows of 48 floats, +2 DWORD gap every 16 DWORDs
// (TDM pad_interval=16DW, pad_amount=2DW) -> off(r,c) = 54r + c + 2*(c>>4).
// Even stride keeps 8B alignment for b64 loads; 54 mod 64 spreads banks.
#define RSTRIDE 54
#define REGION_FLOATS (48 * RSTRIDE + 8)
__device__ __forceinline__ int gidx(int r, int c) {
    return r * RSTRIDE + c + ((c >> 4) << 1);
}

// band(k, m) = 1 if window position k contributes to output m (k in [m+1, m+15])
__device__ __forceinline__ float bandv(int k, int m) {
    return (k > m && k <= m + 15) ? 1.0f : 0.0f;
}

#if HAVE_TDM
typedef unsigned int u32x4 __attribute__((ext_vector_type(4)));
typedef int i32x4 __attribute__((ext_vector_type(4)));
typedef int i32x8 __attribute__((ext_vector_type(8)));

// DMA a 48x48 f32 tile (tile origin = gptr) into LDS at byte offset lds_off.
// rem_w/rem_h = remaining tensor extent from the tile origin; OOB reads return 0,
// which matches the zero-padding of the box filters. LDS padding: 2 DWORDs per 16.
__device__ __forceinline__ void tdm_load_tile48(const void* gptr, unsigned lds_off,
                                                unsigned rem_w, unsigned rem_h) {
    unsigned long long ga = (unsigned long long)(uintptr_t)gptr;
    u32x4 g0;
    g0[0] = 1u;                                   // count=1, user mode, no gather
    g0[1] = lds_off;                              // lds_addr (bytes)
    g0[2] = (unsigned)ga;                         // global_addr[31:0]
    g0[3] = (unsigned)((ga >> 32) & 0x01ffffffu)  // global_addr[56:32]
            | (2u << 30);                         // type=2 (image)
    i32x8 g1;
    g1[0] = (int)((2u << 16)     // data_size = 4 bytes
                | (1u << 20)     // pad_enable
                | (3u << 22)     // pad_interval: 16 DWORDs
                | (1u << 25));   // pad_amount: 2 DWORDs
    g1[1] = (int)((rem_w & 0xffffu) << 16);                       // tensor_dim0[15:0]
    g1[2] = (int)(((rem_w >> 16) & 0xffffu)
                | ((rem_h & 0xffffu) << 16));                     // td0[31:16] | td1[15:0]
    g1[3] = (int)(((rem_h >> 16) & 0xffffu) | (48u << 16));       // td1[31:16] | tile_dim0=48
    g1[4] = 48;                                                   // tile_dim1=48, tile_dim2=0
    g1[5] = WW;                                                   // tensor_dim0_stride[31:0]
    g1[6] = 0;
    g1[7] = 0;
    i32x4 z4 = {};
#if defined(__clang_major__) && __clang_major__ >= 23
    i32x8 z8 = {};
    __builtin_amdgcn_tensor_load_to_lds(g0, g1, z4, z4, z8, 0);
#else
    __builtin_amdgcn_tensor_load_to_lds(g0, g1, z4, z4, 0);
#endif
}
#endif  // HAVE_TDM

// ---------------- Kernel 1: channel min + channel mean ----------------
__global__ __launch_bounds__(256) void prep_kernel(const float* __restrict__ smoky,
                                                   float* __restrict__ dcmin,
                                                   float* __restrict__ guide) {
    int idx = blockIdx.x * 256 + threadIdx.x;
    if (idx >= NBATCH * HWSZ) return;
    int b = idx >> 18;
    int r = idx & (HWSZ - 1);
    const float* p = smoky + (size_t)b * IMG3 + r;
    float s0 = p[0], s1 = p[HWSZ], s2 = p[2 * HWSZ];
    dcmin[idx] = fminf(s0, fminf(s1, s2));
    guide[idx] = (s0 + s1 + s2) * (1.0f / 3.0f);
}

// ---------------- Kernel 2: 15x15 min-pool, replicate pad ----------------
__global__ __launch_bounds__(128) void minpool_kernel(const float* __restrict__ src,
                                                      float* __restrict__ dst) {
    __shared__ float sIn[REGION_FLOATS];
    __shared__ float sMid[48][40];
    int b = blockIdx.x >> 8;
    int t = blockIdx.x & 255;
    int y0 = (t >> 4) << 5;
    int x0 = (t & 15) << 5;
    int tid = threadIdx.x;
    const float* sp = src + (size_t)b * HWSZ;
    int xs = x0 - 8, ys = y0 - 8;
    bool tdm_done = false;
#if HAVE_TDM
    // fully interior tiles: no clamping anywhere -> plain DMA copy
    if (xs >= 0 && ys >= 0 && xs + 48 <= WW && ys + 48 <= HH) {
        tdm_done = true;
        if (__builtin_amdgcn_readfirstlane((int)(threadIdx.x >> 5)) == 0) {
            tdm_load_tile48(sp + (size_t)ys * WW + xs,
                            (unsigned)(uintptr_t)(void*)sIn,
                            (unsigned)(WW - xs), (unsigned)(HH - ys));
            __builtin_amdgcn_s_wait_tensorcnt(0);
        }
    }
#endif
    if (!tdm_done) {
        for (int i = tid; i < 48 * 48; i += 128) {
            int rr = i / 48, cc = i % 48;
            int y = min(max(y0 + rr - 8, 0), HH - 1);
            int x = min(max(x0 + cc - 8, 0), WW - 1);
            sIn[gidx(rr, cc)] = sp[y * WW + x];
        }
    }
    __syncthreads();
    for (int i = tid; i < 48 * 32; i += 128) {
        int rr = i >> 5, c = i & 31;
        float m = sIn[gidx(rr, c + 1)];
        #pragma unroll
        for (int k = 2; k <= 15; ++k) m = fminf(m, sIn[gidx(rr, c + k)]);
        sMid[rr][c] = m;
    }
    __syncthreads();
    float* dp = dst + (size_t)b * HWSZ;
    for (int i = tid; i < 32 * 32; i += 128) {
        int r = i >> 5, c = i & 31;
        float m = sMid[r + 1][c];
        #pragma unroll
        for (int k = 2; k <= 15; ++k) m = fminf(m, sMid[r + k][c]);
        dp[(y0 + r) * WW + (x0 + c)] = m;
    }
}

// ---------------- Kernel 3: guided filter stage 1 (means -> a,b) ----------------
__global__ __launch_bounds__(128) void gf_stage1_kernel(const float* __restrict__ guide,
                                                        const float* __restrict__ dcc,
                                                        float* __restrict__ aplane,
                                                        float* __restrict__ bplane) {
    __shared__ float gIn[REGION_FLOATS];
    __shared__ float pIn[REGION_FLOATS];
    __shared__ float hs[48][40];
    __shared__ float box[4][32][32];
    int b = blockIdx.x >> 8;
    int t = blockIdx.x & 255;
    int y0 = (t >> 4) << 5;
    int x0 = (t & 15) << 5;
    int tid = threadIdx.x;
    const float* gp = guide + (size_t)b * HWSZ;
    const float* pp = dcc + (size_t)b * HWSZ;
    int xs = x0 - 8, ys = y0 - 8;
    bool tdm_done = false;
#if HAVE_TDM
    // interior-origin tiles: right/bottom OOB returns zeros == zero padding
    if (xs >= 0 && ys >= 0) {
        tdm_done = true;
        if (__builtin_amdgcn_readfirstlane((int)(threadIdx.x >> 5)) == 0) {
            tdm_load_tile48(gp + (size_t)ys * WW + xs,
                            (unsigned)(uintptr_t)(void*)gIn,
                            (unsigned)(WW - xs), (unsigned)(HH - ys));
            tdm_load_tile48(pp + (size_t)ys * WW + xs,
                            (unsigned)(uintptr_t)(void*)pIn,
                            (unsigned)(WW - xs), (unsigned)(HH - ys));
            __builtin_amdgcn_s_wait_tensorcnt(0);
        }
    }
#endif
    if (!tdm_done) {
        for (int i = tid; i < 48 * 48; i += 128) {
            int rr = i / 48, cc = i % 48;
            int y = y0 + rr - 8, x = x0 + cc - 8;
            bool inb = (y >= 0 && y < HH && x >= 0 && x < WW);
            int off = inb ? (y * WW + x) : 0;
            gIn[gidx(rr, cc)] = inb ? gp[off] : 0.0f;
            pIn[gidx(rr, cc)] = inb ? pp[off] : 0.0f;
        }
    }
    __syncthreads();
    int wv = tid >> 5;
    int lane = tid & 31;
    int mrow = lane & 15;
    int half = lane >> 4;
    for (int q = 0; q < 4; ++q) {
#if USE_WMMA
        // horizontal: 6 jobs = 3 row-groups x 2 col-tiles; one 16x16 per wave
        for (int job = wv; job < 6; job += 4) {
            int g = job >> 1, ct = job & 1;
            v8f acc = {};
            #pragma unroll
            for (int j = 0; j < 8; ++j) {
                int cc = (ct << 4) + (j << 2) + (half << 1);
                int rrow = (g << 4) + mrow;
                v2f vg = *(const v2f*)&gIn[gidx(rrow, cc)];
                v2f vp = *(const v2f*)&pIn[gidx(rrow, cc)];
                v2f A = (q == 0) ? vg : (q == 1) ? vp : (q == 2) ? (v2f)(vg * vp) : (v2f)(vg * vg);
                int k0 = (j << 2) + (half << 1);
                v2f Bm = { bandv(k0, mrow), bandv(k0 + 1, mrow) };
                acc = __builtin_amdgcn_wmma_f32_16x16x4_f32(false, A, false, Bm,
                                                            (short)0, acc, false, false);
            }
            #pragma unroll
            for (int i = 0; i < 8; ++i)
                hs[(g << 4) + i + (half << 3)][(ct << 4) + mrow] = acc[i];
        }
        __syncthreads();
        // vertical: 4 jobs, one per wave
        {
            int rt = wv >> 1, ct = wv & 1;
            v8f acc = {};
            #pragma unroll
            for (int j = 0; j < 8; ++j) {
                int k0 = (j << 2) + (half << 1);
                v2f A = { bandv(k0, mrow), bandv(k0 + 1, mrow) };
                int col = (ct << 4) + mrow;
                v2f Bm = { hs[(rt << 4) + k0][col], hs[(rt << 4) + k0 + 1][col] };
                acc = __builtin_amdgcn_wmma_f32_16x16x4_f32(false, A, false, Bm,
                                                            (short)0, acc, false, false);
            }
            #pragma unroll
            for (int i = 0; i < 8; ++i)
                box[q][(rt << 4) + i + (half << 3)][(ct << 4) + mrow] = acc[i] * INV225;
        }
        __syncthreads();
#else
        for (int i = tid; i < 48 * 32; i += 128) {
            int rr = i >> 5, c = i & 31;
            float s = 0.0f;
            for (int k = 1; k <= 15; ++k) {
                float gv = gIn[gidx(rr, c + k)], pv = pIn[gidx(rr, c + k)];
                s += (q == 0) ? gv : (q == 1) ? pv : (q == 2) ? gv * pv : gv * gv;
            }
            hs[rr][c] = s;
        }
        __syncthreads();
        for (int i = tid; i < 1024; i += 128) {
            int r = i >> 5, c = i & 31;
            float s = 0.0f;
            for (int k = 1; k <= 15; ++k) s += hs[r + k][c];
            box[q][r][c] = s * INV225;
        }
        __syncthreads();
#endif
    }
    float* ap = aplane + (size_t)b * HWSZ;
    float* bp = bplane + (size_t)b * HWSZ;
    for (int i = tid; i < 1024; i += 128) {
        int r = i >> 5, c = i & 31;
        float mI = box[0][r][c], mp = box[1][r][c];
        float mIp = box[2][r][c], mII = box[3][r][c];
        float cov = mIp - mI * mp;
        float var = mII - mI * mI;
        float a = cov / (var + EPS_F);
        ap[(y0 + r) * WW + (x0 + c)] = a;
        bp[(y0 + r) * WW + (x0 + c)] = mp - a * mI;
    }
}

// ---------------- Kernel 4: stage 2 (mean_a, mean_b) + final combine ----------------
__global__ __launch_bounds__(128) void gf_stage2_kernel(const float* __restrict__ aplane,
                                                        const float* __restrict__ bplane,
                                                        const float* __restrict__ guide,
                                                        const float* __restrict__ smoky,
                                                        const float* __restrict__ rho,
                                                        float* __restrict__ out) {
    __shared__ float aIn[REGION_FLOATS];
    __shared__ float bIn[REGION_FLOATS];
    __shared__ float hs[48][40];
    __shared__ float box[2][32][32];
    int b = blockIdx.x >> 8;
    int t = blockIdx.x & 255;
    int y0 = (t >> 4) << 5;
    int x0 = (t & 15) << 5;
    int tid = threadIdx.x;
    // prefetch combine-stage global inputs early (global_prefetch_b8)
    for (int i = tid; i < 1024; i += 128) {
        int r = i >> 5, c = i & 31;
        size_t base = (size_t)b * IMG3 + (size_t)(y0 + r) * WW + (x0 + c);
        __builtin_prefetch(&smoky[base], 0, 0);
        __builtin_prefetch(&rho[base], 0, 0);
        __builtin_prefetch(&smoky[base + 2 * HWSZ], 0, 0);
        __builtin_prefetch(&rho[base + 2 * HWSZ], 0, 0);
    }
    const float* apg = aplane + (size_t)b * HWSZ;
    const float* bpg = bplane + (size_t)b * HWSZ;
    int xs = x0 - 8, ys = y0 - 8;
    bool tdm_done = false;
#if HAVE_TDM
    if (xs >= 0 && ys >= 0) {
        tdm_done = true;
        if (__builtin_amdgcn_readfirstlane((int)(threadIdx.x >> 5)) == 0) {
            tdm_load_tile48(apg + (size_t)ys * WW + xs,
                            (unsigned)(uintptr_t)(void*)aIn,
                            (unsigned)(WW - xs), (unsigned)(HH - ys));
            tdm_load_tile48(bpg + (size_t)ys * WW + xs,
                            (unsigned)(uintptr_t)(void*)bIn,
                            (unsigned)(WW - xs), (unsigned)(HH - ys));
            __builtin_amdgcn_s_wait_tensorcnt(0);
        }
    }
#endif
    if (!tdm_done) {
        for (int i = tid; i < 48 * 48; i += 128) {
            int rr = i / 48, cc = i % 48;
            int y = y0 + rr - 8, x = x0 + cc - 8;
            bool inb = (y >= 0 && y < HH && x >= 0 && x < WW);
            int off = inb ? (y * WW + x) : 0;
            aIn[gidx(rr, cc)] = inb ? apg[off] : 0.0f;
            bIn[gidx(rr, cc)] = inb ? bpg[off] : 0.0f;
        }
    }
    __syncthreads();
    int wv = tid >> 5;
    int lane = tid & 31;
    int mrow = lane & 15;
    int half = lane >> 4;
    for (int q = 0; q < 2; ++q) {
#if USE_WMMA
        for (int job = wv; job < 6; job += 4) {
            int g = job >> 1, ct = job & 1;
            v8f acc = {};
            #pragma unroll
            for (int j = 0; j < 8; ++j) {
                int cc = (ct << 4) + (j << 2) + (half << 1);
                int rrow = (g << 4) + mrow;
                v2f A = (q == 0) ? *(const v2f*)&aIn[gidx(rrow, cc)]
                                 : *(const v2f*)&bIn[gidx(rrow, cc)];
                int k0 = (j << 2) + (half << 1);
                v2f Bm = { bandv(k0, mrow), bandv(k0 + 1, mrow) };
                acc = __builtin_amdgcn_wmma_f32_16x16x4_f32(false, A, false, Bm,
                                                            (short)0, acc, false, false);
            }
            #pragma unroll
            for (int i = 0; i < 8; ++i)
                hs[(g << 4) + i + (half << 3)][(ct << 4) + mrow] = acc[i];
        }
        __syncthreads();
        {
            int rt = wv >> 1, ct = wv & 1;
            v8f acc = {};
            #pragma unroll
            for (int j = 0; j < 8; ++j) {
                int k0 = (j << 2) + (half << 1);
                v2f A = { bandv(k0, mrow), bandv(k0 + 1, mrow) };
                int col = (ct << 4) + mrow;
                v2f Bm = { hs[(rt << 4) + k0][col], hs[(rt << 4) + k0 + 1][col] };
                acc = __builtin_amdgcn_wmma_f32_16x16x4_f32(false, A, false, Bm,
                                                            (short)0, acc, false, false);
            }
            #pragma unroll
            for (int i = 0; i < 8; ++i)
                box[q][(rt << 4) + i + (half << 3)][(ct << 4) + mrow] = acc[i] * INV225;
        }
        __syncthreads();
#else
        for (int i = tid; i < 48 * 32; i += 128) {
            int rr = i >> 5, c = i & 31;
            float s = 0.0f;
            for (int k = 1; k <= 15; ++k)
                s += (q == 0) ? aIn[gidx(rr, c + k)] : bIn[gidx(rr, c + k)];
            hs[rr][c] = s;
        }
        __syncthreads();
        for (int i = tid; i < 1024; i += 128) {
            int r = i >> 5, c = i & 31;
            float s = 0.0f;
            for (int k = 1; k <= 15; ++k) s += hs[r + k][c];
            box[q][r][c] = s * INV225;
        }
        __syncthreads();
#endif
    }
    const float* gp = guide + (size_t)b * HWSZ;
    for (int i = tid; i < 1024; i += 128) {
        int r = i >> 5, c = i & 31;
        int pix = (y0 + r) * WW + (x0 + c);
        float g = gp[pix];
        float dcr = box[0][r][c] * g + box[1][r][c];
        float scale = (ETA_F + dcr) * (1.0f / (1.0f + ETA_F));
        size_t base = (size_t)b * IMG3 + pix;
        #pragma unroll
        for (int ch = 0; ch < 3; ++ch) {
            size_t off = base + (size_t)ch * HWSZ;
            out[off] = smoky[off] - scale * (1.0f - rho[off]);
        }
    }
}

extern "C" void kernel_launch(void* const* d_in, const int* in_sizes, int n_in,
                              void* d_out, int out_size, void* d_ws, size_t ws_size,
                              hipStream_t stream) {
    const float* smoky = (const float*)d_in[0];
    const float* rho   = (const float*)d_in[1];
    float* out = (float*)d_out;
    float* ws  = (float*)d_ws;
    const size_t P = (size_t)NBATCH * HWSZ;
    float* guide    = ws;
    float* dcmin    = ws + P;
    float* dccoarse = ws + 2 * P;
    float* aplane   = ws + 3 * P;
    float* bplane   = ws + 4 * P;

    int total = NBATCH * HWSZ;
    prep_kernel<<<(total + 255) / 256, 256, 0, stream>>>(smoky, dcmin, guide);

    int tiles = NBATCH * (HH / 32) * (WW / 32);  // 4096 workgroups
    minpool_kernel<<<tiles, 128, 0, stream>>>(dcmin, dccoarse);
    gf_stage1_kernel<<<tiles, 128, 0, stream>>>(guide, dccoarse, aplane, bplane);
    gf_stage2_kernel<<<tiles, 128, 0, stream>>>(aplane, bplane, guide, smoky, rho, out);
}